// GOT_55817394979533
// MI455X (gfx1250) — compile-verified
//
#include <hip/hip_runtime.h>
#include <math.h>

// ---------------- problem constants (from reference) ----------------
constexpr int   kBS = 8;
constexpr int   kD  = 256;
constexpr int   kN  = 512;           // n == m == 512
constexpr float kLam = 0.1f;

typedef __attribute__((ext_vector_type(2))) float v2f;
typedef __attribute__((ext_vector_type(8))) float v8f;

// ---- monotonic float<->uint mapping for atomic min/max over floats ----
__device__ __forceinline__ unsigned got_f2ord(float f) {
  unsigned u = __float_as_uint(f);
  return (u & 0x80000000u) ? ~u : (u | 0x80000000u);
}
__device__ __forceinline__ float got_ord2f(unsigned u) {
  return __uint_as_float((u & 0x80000000u) ? (u & 0x7fffffffu) : ~u);
}

__device__ __forceinline__ float got_block_reduce(float v, float* red) {
  const int t = threadIdx.x;
  red[t] = v;
  __syncthreads();
  for (int o = 128; o > 0; o >>= 1) {
    if (t < o) red[t] += red[t + o];
    __syncthreads();
  }
  return red[0];
}

// Load a K-pair fragment element: {p[k*S], p[(k+1)*S]}; stride-1 becomes one b64 load.
template <int S>
__device__ __forceinline__ v2f got_load2(const float* __restrict__ p, long k) {
  if constexpr (S == 1) {
    return *(const v2f*)(p + k);          // k is even -> 8B aligned
  } else {
    v2f r;
    r.x = p[k * S];
    r.y = p[(k + 1) * S];
    return r;
  }
}

// ---------------- normalize columns over D (axis=1) ----------------
__global__ void got_normalize(const float* __restrict__ in, float* __restrict__ out) {
  const int b = blockIdx.y;
  const int l = blockIdx.x * blockDim.x + threadIdx.x;   // 0..kN-1
  const float* p = in + (long)b * kD * kN + l;
  float s = 0.f;
  for (int d = 0; d < kD; ++d) { float v = p[(long)d * kN]; s += v * v; }
  const float inv = 1.0f / (sqrtf(s) + 1e-12f);
  float* o = out + (long)b * kD * kN + l;
  for (int d = 0; d < kD; ++d) o[(long)d * kN] = p[(long)d * kN] * inv;
}

// ---------------- conv1d, SAME padding, kernel width 3 ----------------
__global__ void got_conv1d(const float* __restrict__ X, const float* __restrict__ W,
                           const float* __restrict__ bias, float* __restrict__ out) {
  const int b  = blockIdx.z;
  const int co = blockIdx.y;
  const int l  = blockIdx.x * blockDim.x + threadIdx.x;
  const float* Xb = X + (long)b * kD * kN;
  const float* Wr = W + (long)co * kD * 3;
  float s = bias[co];
  for (int ci = 0; ci < kD; ++ci) {
    const float* xr = Xb + (long)ci * kN;
    const float w0 = Wr[ci * 3 + 0], w1 = Wr[ci * 3 + 1], w2 = Wr[ci * 3 + 2];
    const float xm = (l > 0)      ? xr[l - 1] : 0.f;
    const float x0 = xr[l];
    const float xp = (l < kN - 1) ? xr[l + 1] : 0.f;
    s += w0 * xm + w1 * x0 + w2 * xp;
  }
  out[((long)b * kD + co) * kN + l] = s;
}

// ---------------- strided batched WMMA GEMM, 16x64 tile per wave ----------------
// C[b][i][j] = sum_k A(b,k,i) * B(b,k,j); A elem at k*SAK + i*SAI, B at k*SBK + j*SBJ.
// One wave computes a 16x64 output tile: 1 A-fragment reused across 4 B-fragments.
// EPI 0: cosine dist (1-acc) + global min/max atomics into mm[0..1]
// EPI 1: plain store
// EPI 2: A=exp(-2*(lam*CW + (1-lam)*(rA[i]+rB[j]-2*acc))), C2=1 (IPOT init)
// EPI 3: final C_gamma = rA[i]+rB[j]-2*acc (no lambda mix, per reference)
template <int EPI, int SAK, int SAI, int SBK, int SBJ, int K>
__global__ __launch_bounds__(32) void got_gemm(
    const float* __restrict__ A, long abat,
    const float* __restrict__ B, long bbat,
    float* __restrict__ C, float* __restrict__ C2, const float* __restrict__ CW,
    const float* __restrict__ rA, const float* __restrict__ rB,
    unsigned* __restrict__ mm) {
  const int b    = blockIdx.z;
  const int i0   = blockIdx.y << 4;
  const int j0   = blockIdx.x << 6;  // 64 columns per wave
  const int lane = threadIdx.x;      // 0..31 (wave32)
  const int half = lane >> 4;        // 0/1
  const int lm   = lane & 15;

  const float* Ap  = A + (long)b * abat + (long)(i0 + lm) * SAI;
  const float* Bp0 = B + (long)b * bbat + (long)(j0 + lm) * SBJ;
  const float* Bp1 = Bp0 + 16L * SBJ;
  const float* Bp2 = Bp0 + 32L * SBJ;
  const float* Bp3 = Bp0 + 48L * SBJ;

  v8f acc0 = {}, acc1 = {}, acc2 = {}, acc3 = {};
#pragma unroll 4
  for (int k0 = 0; k0 < K; k0 += 4) {
    // f32 16x16x4 frag: lanes 0-15 hold K=k0,k0+1; lanes 16-31 hold K=k0+2,k0+3
    const long ka = k0 + half * 2;
    const v2f af = got_load2<SAK>(Ap, ka);
    const v2f b0 = got_load2<SBK>(Bp0, ka);
    const v2f b1 = got_load2<SBK>(Bp1, ka);
    const v2f b2 = got_load2<SBK>(Bp2, ka);
    const v2f b3 = got_load2<SBK>(Bp3, ka);
    acc0 = __builtin_amdgcn_wmma_f32_16x16x4_f32(false, af, false, b0, (short)0, acc0, false, false);
    acc1 = __builtin_amdgcn_wmma_f32_16x16x4_f32(false, af, false, b1, (short)0, acc1, false, false);
    acc2 = __builtin_amdgcn_wmma_f32_16x16x4_f32(false, af, false, b2, (short)0, acc2, false, false);
    acc3 = __builtin_amdgcn_wmma_f32_16x16x4_f32(false, af, false, b3, (short)0, acc3, false, false);
  }

  float lmin = 3.4e38f, lmax = -3.4e38f;
  auto epilogue = [&](const v8f& acc, int jb) {
    const int col = jb + lm;
#pragma unroll
    for (int r = 0; r < 8; ++r) {
      // C/D layout: VGPR r -> M=r (lanes 0-15), M=r+8 (lanes 16-31); N=lane%16
      const int  row = i0 + half * 8 + r;
      const long idx = ((long)b * kN + row) * kN + col;
      const float v  = acc[r];
      if (EPI == 0) {
        const float cd = 1.0f - v;
        C[idx] = cd;
        lmin = fminf(lmin, cd);
        lmax = fmaxf(lmax, cd);
      } else if (EPI == 1) {
        C[idx] = v;
      } else if (EPI == 2) {
        const float cg = kLam * CW[idx] +
            (1.0f - kLam) * (rA[b * kN + row] + rB[b * kN + col] - 2.0f * v);
        C[idx]  = expf(-2.0f * cg);   // exp(-C/beta), beta = 0.5
        C2[idx] = 1.0f;               // T0 = ones
      } else {                        // EPI == 3
        C[idx] = rA[b * kN + row] + rB[b * kN + col] - 2.0f * v;
      }
    }
  };
  epilogue(acc0, j0);
  epilogue(acc1, j0 + 16);
  epilogue(acc2, j0 + 32);
  epilogue(acc3, j0 + 48);

  if (EPI == 0) {
#pragma unroll
    for (int off = 16; off > 0; off >>= 1) {
      lmin = fminf(lmin, __shfl_xor(lmin, off, 32));
      lmax = fmaxf(lmax, __shfl_xor(lmax, off, 32));
    }
    if (lane == 0) {
      atomicMin(&mm[0], got_f2ord(lmin));
      atomicMax(&mm[1], got_f2ord(lmax));
    }
  }
}

// ---------------- threshold + relu (+ optional transpose) ----------------
__global__ void got_threshold(const float* __restrict__ cd, const unsigned* __restrict__ mm,
                              float* __restrict__ out, int transpose) {
  const long idx = (long)blockIdx.x * blockDim.x + threadIdx.x;
  const float mn  = got_ord2f(mm[0]);
  const float mx  = got_ord2f(mm[1]);
  const float thr = mn + 0.1f * (mx - mn);
  float v = cd[idx] - thr;
  v = v > 0.0f ? v : 0.0f;
  if (transpose) {
    const int  j = (int)(idx & (kN - 1));
    const long t = idx >> 9;
    const int  i = (int)(t & (kN - 1));
    const long b = t >> 9;
    out[(b * kN + j) * kN + i] = v;
  } else {
    out[idx] = v;
  }
}

// rs[b][i] = (1/kN) * sum_k M[b][i][k]^2   (the Cs^2 @ q / Ct^2 @ p vectors)
__global__ void got_rowsq(const float* __restrict__ M, float* __restrict__ out) {
  __shared__ float red[256];
  const int  b    = blockIdx.y, i = blockIdx.x;
  const long base = ((long)b * kN + i) * kN;
  float acc = 0.f;
  for (int j = threadIdx.x; j < kN; j += 256) { const float v = M[base + j]; acc += v * v; }
  const float tot = got_block_reduce(acc, red);
  if (threadIdx.x == 0) out[b * kN + i] = tot * (1.0f / (float)kN);
}

// ---------------- IPOT inner-iteration kernels ----------------
// delta[b][i] = 1 / (n * sum_j A*T*sigma[j])
__global__ void got_ipot_row(const float* __restrict__ A, const float* __restrict__ T,
                             const float* __restrict__ sg, float* __restrict__ dlt) {
  __shared__ float red[256];
  const int  b    = blockIdx.y, i = blockIdx.x;
  const long base = ((long)b * kN + i) * kN;
  const float* s  = sg + b * kN;
  float acc = 0.f;
  for (int j = threadIdx.x; j < kN; j += 256) acc += A[base + j] * T[base + j] * s[j];
  const float tot = got_block_reduce(acc, red);
  if (threadIdx.x == 0) dlt[b * kN + i] = 1.0f / ((float)kN * tot);
}

// sigma[b][j] = 1 / (m * sum_i A*T*delta[i])
__global__ void got_ipot_col(const float* __restrict__ A, const float* __restrict__ T,
                             const float* __restrict__ dlt, float* __restrict__ sg) {
  const int b = blockIdx.y;
  const int j = blockIdx.x * 256 + threadIdx.x;
  const float* d  = dlt + b * kN;
  const long base = (long)b * kN * kN + j;
  float acc = 0.f;
  for (int i = 0; i < kN; ++i) {
    __builtin_prefetch(&A[base + (long)(i + 8) * kN], 0, 1);  // global_prefetch_b8
    __builtin_prefetch(&T[base + (long)(i + 8) * kN], 0, 1);
    acc += A[base + (long)i * kN] * T[base + (long)i * kN] * d[i];
  }
  sg[b * kN + j] = 1.0f / ((float)kN * acc);
}

// T = delta[i] * A * T * sigma[j]
__global__ void got_ipot_upd(const float* __restrict__ A, const float* __restrict__ dlt,
                             const float* __restrict__ sg, float* __restrict__ T) {
  const long idx = (long)blockIdx.x * blockDim.x + threadIdx.x;
  const int  j = (int)(idx & (kN - 1));
  const long t = idx >> 9;
  const int  i = (int)(t & (kN - 1));
  const long b = t >> 9;
  T[idx] = dlt[b * kN + i] * A[idx] * T[idx] * sg[b * kN + j];
}

// out[b] = sum Tg * (lam*CwT + (1-lam)*C_gamma)   (= lam*wd + (1-lam)*gwd)
__global__ void got_final(const float* __restrict__ Cw, const float* __restrict__ Cg,
                          const float* __restrict__ Tg, float* __restrict__ out) {
  __shared__ float red[256];
  const int  b    = blockIdx.x;
  const long base = (long)b * kN * kN;
  float acc = 0.f;
  for (long t = threadIdx.x; t < (long)kN * kN; t += 256)
    acc += Tg[base + t] * (kLam * Cw[base + t] + (1.0f - kLam) * Cg[base + t]);
  const float tot = got_block_reduce(acc, red);
  if (threadIdx.x == 0) out[b] = tot;
}

__global__ void got_fill(float* __restrict__ p, float v, long n) {
  const long idx = (long)blockIdx.x * blockDim.x + threadIdx.x;
  if (idx < n) p[idx] = v;
}

__global__ void got_init_mm(unsigned* __restrict__ mm) {
  mm[0] = 0xFFFFFFFFu;  // min identity in monotonic-uint space
  mm[1] = 0u;           // max identity
}

// ---------------- host orchestration ----------------
extern "C" void kernel_launch(void* const* d_in, const int* in_sizes, int n_in,
                              void* d_out, int out_size, void* d_ws, size_t ws_size,
                              hipStream_t stream) {
  (void)in_sizes; (void)n_in; (void)out_size; (void)ws_size;
  const float* X  = (const float*)d_in[0];
  const float* Y  = (const float*)d_in[1];
  const float* W1 = (const float*)d_in[2];
  const float* b1 = (const float*)d_in[3];
  const float* W2 = (const float*)d_in[4];
  const float* b2 = (const float*)d_in[5];
  float* out = (float*)d_out;

  const size_t SZ_XY = (size_t)kBS * kD * kN;   // 1,048,576 floats
  const size_t SZ_MM = (size_t)kBS * kN * kN;   // 2,097,152 floats
  float* ws  = (float*)d_ws;
  float* Xn  = ws;  ws += SZ_XY;
  float* Yn  = ws;  ws += SZ_XY;
  float* Xi  = ws;  ws += SZ_XY;
  float* Yi  = ws;  ws += SZ_XY;
  float* Cs  = ws;  ws += SZ_MM;
  float* Ct  = ws;  ws += SZ_MM;
  float* Cw  = ws;  ws += SZ_MM;   // Cst_w^T (untransposed relu(cd - thr))
  float* Tg  = ws;  ws += SZ_MM;   // gamma^T; also IPOT plan T
  float* Tmp = ws;  ws += SZ_MM;   // cd scratch / Cs@Tg
  float* Ab  = ws;  ws += SZ_MM;   // exp(-2*C_gamma) / final C_gamma
  float* rsA = ws;  ws += (size_t)kBS * kN;
  float* rsB = ws;  ws += (size_t)kBS * kN;
  float* dlt = ws;  ws += (size_t)kBS * kN;
  float* sg  = ws;  ws += (size_t)kBS * kN;
  unsigned* mm = (unsigned*)ws;

  const dim3 gN(kN / 256, kBS), bN(256);
  const dim3 gC(kN / 128, kD, kBS), bC(128);
  const dim3 gG(kN / 64, kN / 16, kBS), bG(32);   // 16x64 tile per wave
  const long aXY = (long)kD * kN, aMM = (long)kN * kN;
  const int  nEW = (int)(SZ_MM / 256);

  // --- phase A: normalize + conv1d + normalize ---
  got_normalize<<<gN, bN, 0, stream>>>(X, Xn);
  got_normalize<<<gN, bN, 0, stream>>>(Y, Yn);
  got_conv1d<<<gC, bC, 0, stream>>>(X, W1, b1, Xi);
  got_conv1d<<<gC, bC, 0, stream>>>(Y, W2, b2, Yi);
  got_normalize<<<gN, bN, 0, stream>>>(Xi, Xi);   // in-place: column-local
  got_normalize<<<gN, bN, 0, stream>>>(Yi, Yi);

  // --- phase B: cosine grams (A^T B: sak=kN, sai=1; K=kD) -> threshold ---
  auto gram = [&](const float* P, const float* Q, float* dst, int transpose) {
    got_init_mm<<<1, 1, 0, stream>>>(mm);
    got_gemm<0, kN, 1, kN, 1, kD><<<gG, bG, 0, stream>>>(
        P, aXY, Q, aXY, Tmp, nullptr, nullptr, nullptr, nullptr, mm);
    got_threshold<<<nEW, 256, 0, stream>>>(Tmp, mm, dst, transpose);
  };
  gram(Xn, Xn, Cs, 1);
  gram(Yn, Yn, Ct, 1);
  gram(Xi, Yi, Cw, 0);   // untransposed == Cst_w^T

  got_rowsq<<<dim3(kN, kBS), 256, 0, stream>>>(Cs, rsA);
  got_rowsq<<<dim3(kN, kBS), 256, 0, stream>>>(Ct, rsB);
  got_fill<<<nEW, 256, 0, stream>>>(Tg, 1.0f / ((float)kN * (float)kN), (long)SZ_MM);

  // --- phase C: 5 outer iterations of {2 GEMMs + IPOT(20)} ---
  for (int it = 0; it < 5; ++it) {
    // Tmp = Cs @ Tg   (A@B: A(k,i)=Cs[i][k] -> sak=1,sai=kN; B(k,j)=Tg[k][j] -> sbk=kN,sbj=1)
    got_gemm<1, 1, kN, kN, 1, kN><<<gG, bG, 0, stream>>>(
        Cs, aMM, Tg, aMM, Tmp, nullptr, nullptr, nullptr, nullptr, nullptr);
    // Ab = exp(-2*C_gamma), Tg = 1   (A@B^T: B(k,j)=Ct[j][k] -> sbk=1,sbj=kN)
    got_gemm<2, 1, kN, 1, kN, kN><<<gG, bG, 0, stream>>>(
        Tmp, aMM, Ct, aMM, Ab, Tg, Cw, rsA, rsB, nullptr);
    got_fill<<<(kBS * kN) / 256, 256, 0, stream>>>(sg, 1.0f / (float)kN, (long)kBS * kN);
    for (int ot = 0; ot < 20; ++ot) {
      got_ipot_row<<<dim3(kN, kBS), 256, 0, stream>>>(Ab, Tg, sg, dlt);
      got_ipot_col<<<dim3(kN / 256, kBS), 256, 0, stream>>>(Ab, Tg, dlt, sg);
      got_ipot_upd<<<nEW, 256, 0, stream>>>(Ab, dlt, sg, Tg);
    }
  }

  // --- final C_gamma (no lambda mix) + fused traces ---
  got_gemm<1, 1, kN, kN, 1, kN><<<gG, bG, 0, stream>>>(
      Cs, aMM, Tg, aMM, Tmp, nullptr, nullptr, nullptr, nullptr, nullptr);
  got_gemm<3, 1, kN, 1, kN, kN><<<gG, bG, 0, stream>>>(
      Tmp, aMM, Ct, aMM, Ab, nullptr, nullptr, rsA, rsB, nullptr);
  got_final<<<kBS, 256, 0, stream>>>(Cw, Ab, Tg, out);
}